// CustomSparseEncoder_49211735277955
// MI455X (gfx1250) — compile-verified
//
#include <hip/hip_runtime.h>
#include <stdint.h>

// CDNA5 / gfx1250 submanifold sparse conv:
//   per-offset: async-gather rows -> LDS, WMMA f32 16x16x4 GEMM, atomic scatter-add.
//   Weight tile re-packed in LDS as K-pairs so every B fragment is one ds_load_b64.

typedef float v2f __attribute__((ext_vector_type(2)));
typedef float v8f __attribute__((ext_vector_type(8)));

#define TILE_P    128
#define CIN       64
#define COUT      64
#define FSTRIDE   68    // feature LDS row stride (floats): 272B, 16B aligned, conflict-free
#define WTSTRIDE  160   // packed-weight LDS stride per kk (floats): 640B => 32-bank rotation

__device__ __forceinline__ void async_cp16(const void* g, void* l) {
    // global_load_async_to_lds_b128: per-lane 16B global -> LDS, tracked by ASYNCcnt.
    unsigned laddr = (unsigned)(unsigned long long)l;   // low 32 bits of generic = LDS offset
    unsigned long long gaddr = (unsigned long long)g;
    asm volatile("global_load_async_to_lds_b128 %0, %1, off"
                 :: "v"(laddr), "v"(gaddr) : "memory");
}

__global__ __launch_bounds__(256) void bias_init_kernel(
    float* __restrict__ out, const float* __restrict__ bias, int total)
{
    int i = blockIdx.x * blockDim.x + threadIdx.x;
    if (i < total) out[i] = bias[i & (COUT - 1)];
}

__global__ __launch_bounds__(256) void subm_conv_tile_kernel(
    const float*   __restrict__ feat,    // [P, CIN]
    const float*   __restrict__ weight,  // [K, CIN, COUT]
    const int*     __restrict__ pairs,   // [K, 2, P]
    const uint8_t* __restrict__ mask,    // [K, P] (jnp bool -> 1 byte)
    float*         __restrict__ out,     // [P, COUT], pre-initialized with bias
    int P)
{
    __shared__ float fS[TILE_P * FSTRIDE];      // gathered feature rows          (34.0 KB)
    __shared__ float wT[(CIN / 2) * WTSTRIDE];  // packed weight: {w[2kk][c], w[2kk+1][c]} (20 KB)
    __shared__ int   oIdxS[TILE_P];             // output row per pair, -1 = skip

    const int tid  = threadIdx.x;
    const int k    = blockIdx.y;
    const int rowBaseG = blockIdx.x * TILE_P;

    const int*     inIdx  = pairs + (size_t)k * 2 * P;
    const int*     outIdx = inIdx + P;
    const uint8_t* mk     = mask + (size_t)k * P;

    // --- out-index + validity table (masked or OOB pair => -1 => no scatter) ---
    if (tid < TILE_P) {
        int gr = rowBaseG + tid;
        int oi = -1;
        if (gr < P && mk[gr]) oi = outIdx[gr];
        oIdxS[tid] = oi;
    }

    // --- async gather: 128 rows x 256B, 16 lanes per row, one B128 per lane ---
#pragma unroll
    for (int j = 0; j < 8; ++j) {
        int idx = j * 256 + tid;
        int row = idx >> 4;
        int seg = idx & 15;
        int gr  = rowBaseG + row;
        int src = (gr < P) ? inIdx[gr] : 0;        // clamp; row never scattered if OOB
        async_cp16(feat + (size_t)src * CIN + seg * 4,
                   &fS[row * FSTRIDE + seg * 4]);
    }

    // --- weight tile: load + pack K-pairs so B fragments are contiguous b64 ---
    // wT[kk*WTSTRIDE + 2c] = w[2kk][c], w[2kk+1][c]   (kk = 0..31, c = 0..63)
    const float* wk = weight + (size_t)k * CIN * COUT;
#pragma unroll
    for (int j = 0; j < 8; ++j) {
        int i  = j * 256 + tid;
        int kk = i >> 6;       // 0..31
        int c  = i & 63;
        v2f p;
        p.x = wk[(2 * kk)     * COUT + c];
        p.y = wk[(2 * kk + 1) * COUT + c];
        *(v2f*)&wT[kk * WTSTRIDE + c * 2] = p;
    }

    asm volatile("s_wait_asynccnt 0" ::: "memory");
    __syncthreads();

    // --- WMMA: rows [16*wave, 16*wave+16) x full 64 cols, K=64 in steps of 4 ---
    const int lane = tid & 31;
    const int wave = tid >> 5;        // wave32
    const int half = lane >> 4;       // A/B: lanes 16-31 hold K+2/K+3 ; D: M+8
    const int l16  = lane & 15;
    const int mrow = wave * 16;

    const float* aBase = &fS[(mrow + l16) * FSTRIDE];
    const float* bBase = &wT[half * WTSTRIDE + l16 * 2];

    v8f acc0 = {}, acc1 = {}, acc2 = {}, acc3 = {};
#pragma unroll
    for (int k0 = 0; k0 < CIN; k0 += 4) {
        // A 16x4 fragment: lane's row = mrow+l16, K = {ka, ka+1} (contiguous b64)
        v2f a = *(const v2f*)(aBase + k0 + 2 * half);
        // B 4x16 fragments: packed pair {w[ka][c], w[ka+1][c]} -> single b64 each
        const float* brow = bBase + (k0 >> 1) * WTSTRIDE;
        v2f b0 = *(const v2f*)(brow + 0);
        v2f b1 = *(const v2f*)(brow + 32);
        v2f b2 = *(const v2f*)(brow + 64);
        v2f b3 = *(const v2f*)(brow + 96);
        acc0 = __builtin_amdgcn_wmma_f32_16x16x4_f32(false, a, false, b0, (short)0, acc0, false, false);
        acc1 = __builtin_amdgcn_wmma_f32_16x16x4_f32(false, a, false, b1, (short)0, acc1, false, false);
        acc2 = __builtin_amdgcn_wmma_f32_16x16x4_f32(false, a, false, b2, (short)0, acc2, false, false);
        acc3 = __builtin_amdgcn_wmma_f32_16x16x4_f32(false, a, false, b3, (short)0, acc3, false, false);
    }

    // --- scatter-add per D layout: VGPR i holds M = i + 8*half, N = l16 ---
#pragma unroll
    for (int i = 0; i < 8; ++i) {
        int m  = i + 8 * half;
        int oi = oIdxS[mrow + m];
        if (oi >= 0) {
            float* o = out + (size_t)oi * COUT + l16;
            atomicAdd(o +  0, acc0[i]);
            atomicAdd(o + 16, acc1[i]);
            atomicAdd(o + 32, acc2[i]);
            atomicAdd(o + 48, acc3[i]);
        }
    }
}

extern "C" void kernel_launch(void* const* d_in, const int* in_sizes, int n_in,
                              void* d_out, int out_size, void* d_ws, size_t ws_size,
                              hipStream_t stream) {
    const float*   feat   = (const float*)d_in[0];
    const float*   weight = (const float*)d_in[1];
    const float*   bias   = (const float*)d_in[2];
    const int*     pairs  = (const int*)d_in[3];
    const uint8_t* mask   = (const uint8_t*)d_in[4];
    float*         out    = (float*)d_out;

    const int P = in_sizes[0] / CIN;                 // 200000
    const int K = in_sizes[1] / (CIN * COUT);        // 27

    // out = bias (d_out is poisoned before timing; must re-init every call)
    int total = P * COUT;
    bias_init_kernel<<<(total + 255) / 256, 256, 0, stream>>>(out, bias, total);

    dim3 grid((P + TILE_P - 1) / TILE_P, K);
    subm_conv_tile_kernel<<<grid, 256, 0, stream>>>(feat, weight, pairs, mask, out, P);
}